// DarkChannelPriorLoss_4148938407991
// MI455X (gfx1250) — compile-verified
//
#include <hip/hip_runtime.h>
#include <hip/hip_bf16.h>
#include <stdint.h>

// Problem geometry (fixed by the reference): rgb (32,3,512,512) f32
#define DCP_B 32
#define DCP_H 512
#define DCP_W 512
#define RAD   20
#define KW    41
#define NPIX  ((size_t)DCP_B * DCP_H * DCP_W)     // 8388608
#define INV_N (1.0f / 8388608.0f)

// Vertical-pass tile
#define TW 64
#define TH 32

typedef __attribute__((ext_vector_type(2))) float v2f;
typedef __attribute__((ext_vector_type(8))) float v8f;

#if defined(__HIP_DEVICE_COMPILE__) && __has_builtin(__builtin_amdgcn_global_load_async_to_lds_b32)
#define HAVE_ASYNC 1
#else
#define HAVE_ASYNC 0
#endif

#if defined(__HIP_DEVICE_COMPILE__) && __has_builtin(__builtin_amdgcn_wmma_f32_16x16x4_f32)
#define HAVE_WMMA 1
#else
#define HAVE_WMMA 0
#endif

__device__ __forceinline__ void wait_async_zero() {
#if defined(__HIP_DEVICE_COMPILE__) && __has_builtin(__builtin_amdgcn_s_wait_asynccnt)
  __builtin_amdgcn_s_wait_asynccnt(0);
#elif defined(__HIP_DEVICE_COMPILE__)
  asm volatile("s_wait_asynccnt 0" ::: "memory");
#endif
}

// Sum of all 32 lane values of x across the wave.
// WMMA path: A(16x4) holds the 32 partials (VGPR0 slots; VGPR1 = 0),
// B(4x16) = ones  =>  D[m,n] = rowsum(A,m). Summing D's 8 VGPRs gives
// sum over M=0..7 at lanes 0-15 and M=8..15 at lanes 16-31; lane0+lane16
// is the full wave sum. EXEC is all-ones at every call site.
__device__ __forceinline__ float wave_reduce_add(float x) {
#if HAVE_WMMA
  v2f a;  a[0] = x;    a[1] = 0.0f;
  v2f bm; bm[0] = 1.0f; bm[1] = 1.0f;
  v8f c = {};
  v8f d = __builtin_amdgcn_wmma_f32_16x16x4_f32(
      /*neg_a=*/false, a, /*neg_b=*/false, bm,
      /*c_mod=*/(short)0, c, /*reuse_a=*/false, /*reuse_b=*/false);
  float s = d[0] + d[1] + d[2] + d[3] + d[4] + d[5] + d[6] + d[7];
  return __shfl(s, 0, 32) + __shfl(s, 16, 32);
#else
  for (int off = 16; off > 0; off >>= 1) x += __shfl_xor(x, off, 32);
  return x;
#endif
}

__device__ __forceinline__ void block_reduce_atomic_add(float ps, float* loss) {
  __shared__ float wsum[8];
  float ws = wave_reduce_add(ps);
  const int lane = threadIdx.x & 31;
  const int wid  = threadIdx.x >> 5;
  if (lane == 0) wsum[wid] = ws;
  __syncthreads();
  if (threadIdx.x == 0) {
    float tot = 0.0f;
    const int nw = (blockDim.x + 31) >> 5;
    for (int i = 0; i < nw; ++i) tot += wsum[i];
    atomicAdd(loss, tot);
  }
}

__global__ void k_zero(float* loss) { *loss = 0.0f; }

// Pass 1: channel-min + horizontal 41-tap min over one full image row.
// grid = B*H blocks, 512 threads.
__global__ __launch_bounds__(512) void k_hmin(const float* __restrict__ rgb,
                                              float* __restrict__ rowmin) {
  __shared__ float s[DCP_W + 2 * RAD];  // +inf halo so the 41-tap loop is branch-free
  const int t = threadIdx.x;
  const int h = blockIdx.x & (DCP_H - 1);
  const int b = blockIdx.x >> 9;
  if (t < RAD) {
    s[t] = __builtin_inff();
    s[DCP_W + RAD + t] = __builtin_inff();
  }
  const size_t plane = (size_t)DCP_H * DCP_W;
  const float* row0 = rgb + (size_t)b * 3 * plane + (size_t)h * DCP_W;
  const float m0 = row0[t];
  const float m1 = row0[plane + t];
  const float m2 = row0[2 * plane + t];
  s[RAD + t] = fminf(m0, fminf(m1, m2));
  __syncthreads();
  float m = s[t];
#pragma unroll
  for (int i = 1; i < KW; ++i) m = fminf(m, s[t + i]);
  rowmin[((size_t)b * DCP_H + h) * DCP_W + t] = m;
}

// Pass 2: vertical 41-tap min on a 64x32 output tile (64x72 halo tile in LDS,
// staged with async global->LDS when available), write dcp, accumulate mean.
// grid = (W/TW, H/TH, B), 256 threads.
__global__ __launch_bounds__(256) void k_vmin(const float* __restrict__ rowmin,
                                              float* __restrict__ dcp,
                                              float* __restrict__ loss) {
  __shared__ float tile[(TH + 2 * RAD) * TW];  // 72*64 f32 = 18 KB
  const int tx = threadIdx.x & (TW - 1);
  const int ty = threadIdx.x >> 6;  // 0..3
  const int w0 = blockIdx.x * TW;
  const int h0 = blockIdx.y * TH;
  const int b  = blockIdx.z;
  const float* src = rowmin + (size_t)b * DCP_H * DCP_W;

  for (int rr = ty; rr < TH + 2 * RAD; rr += 4) {
    const int hh = h0 - RAD + rr;
    float* dst = &tile[rr * TW + tx];
    if (hh >= 0 && hh < DCP_H) {
#if HAVE_ASYNC
      __builtin_amdgcn_global_load_async_to_lds_b32(
          (__attribute__((address_space(1))) int*)(src + (size_t)hh * DCP_W + w0 + tx),
          (__attribute__((address_space(3))) int*)dst,
          /*imm offset=*/0, /*cpol=*/0);
#else
      *dst = src[(size_t)hh * DCP_W + w0 + tx];
#endif
    } else {
      *dst = __builtin_inff();
    }
  }
#if HAVE_ASYNC
  wait_async_zero();
#endif
  __syncthreads();

  float ps = 0.0f;
  float* out = dcp + ((size_t)b * DCP_H + h0) * DCP_W + w0;
  for (int orow = ty; orow < TH; orow += 4) {
    float m = tile[orow * TW + tx];
#pragma unroll
    for (int i = 1; i < KW; ++i) m = fminf(m, tile[(orow + i) * TW + tx]);
    const float v = fabsf(m);
    out[(size_t)orow * DCP_W + tx] = v;
    ps += v;
  }

  block_reduce_atomic_add(ps * INV_N, loss);
}

// Fallback (only if ws_size is too small): direct 3x41x41 min per pixel.
__global__ __launch_bounds__(256) void k_direct(const float* __restrict__ rgb,
                                                float* __restrict__ dcp,
                                                float* __restrict__ loss) {
  const size_t idx = (size_t)blockIdx.x * 256 + threadIdx.x;
  const int w = (int)(idx & (DCP_W - 1));
  const int h = (int)((idx >> 9) & (DCP_H - 1));
  const int b = (int)(idx >> 18);
  const size_t plane = (size_t)DCP_H * DCP_W;
  const float* base = rgb + (size_t)b * 3 * plane;
  const int hlo = (h - RAD < 0) ? 0 : h - RAD;
  const int hhi = (h + RAD > DCP_H - 1) ? DCP_H - 1 : h + RAD;
  const int wlo = (w - RAD < 0) ? 0 : w - RAD;
  const int whi = (w + RAD > DCP_W - 1) ? DCP_W - 1 : w + RAD;
  float m = __builtin_inff();
  for (int c = 0; c < 3; ++c) {
    const float* p = base + (size_t)c * plane;
    for (int hh = hlo; hh <= hhi; ++hh) {
      const float* r = p + (size_t)hh * DCP_W;
      for (int ww = wlo; ww <= whi; ++ww) m = fminf(m, r[ww]);
    }
  }
  const float v = fabsf(m);
  dcp[idx] = v;
  block_reduce_atomic_add(v * INV_N, loss);
}

extern "C" void kernel_launch(void* const* d_in, const int* in_sizes, int n_in,
                              void* d_out, int out_size, void* d_ws, size_t ws_size,
                              hipStream_t stream) {
  (void)in_sizes; (void)n_in; (void)out_size;
  const float* rgb = (const float*)d_in[0];
  float* out  = (float*)d_out;
  float* loss = out;       // d_out[0] = loss scalar
  float* dcp  = out + 1;   // d_out[1..] = dcp (32,1,512,512) flat

  k_zero<<<1, 1, 0, stream>>>(loss);

  const size_t need = NPIX * sizeof(float);
  if (d_ws != nullptr && ws_size >= need) {
    float* rowmin = (float*)d_ws;
    k_hmin<<<dim3(DCP_B * DCP_H), 512, 0, stream>>>(rgb, rowmin);
    k_vmin<<<dim3(DCP_W / TW, DCP_H / TH, DCP_B), 256, 0, stream>>>(rowmin, dcp, loss);
  } else {
    k_direct<<<dim3((unsigned)(NPIX / 256)), 256, 0, stream>>>(rgb, dcp, loss);
  }
}